// LightDeepFilterNet_47210280517691
// MI455X (gfx1250) — compile-verified
//
#include <hip/hip_runtime.h>

// ---------------- problem constants (match reference) ----------------
#define NFREQ 96
#define FTOT  481
#define FS    5
#define PADB  4      // FRAME_SIZE - 1 - LOOKAHEAD
#define TLEN  4000
#define BATCH 16
#define TT    8      // t-steps per filter block
#define NPASS (FTOT - NFREQ)                    // 385 passthrough complexes per row
#define NCOPY (BATCH * TLEN * NPASS)            // 24,640,000 float2 elements
#define CPB_THREADS 256
#define CPB_K 8

// Detect gfx1250 async global<->LDS builtins at device-compile time.
#if defined(__HIP_DEVICE_COMPILE__) && \
    __has_builtin(__builtin_amdgcn_global_load_async_to_lds_b64) && \
    __has_builtin(__builtin_amdgcn_global_store_async_from_lds_b64)
#define USE_ASYNC_COPY 1
#else
#define USE_ASYNC_COPY 0
#endif

struct f2 { float x, y; };

#if USE_ASYNC_COPY
// b64 async payload type per the builtin's signature: int __attribute__((vector_size(8)))
typedef int v2i __attribute__((vector_size(2 * sizeof(int))));
typedef __attribute__((address_space(1))) v2i* gptr;   // global side
typedef __attribute__((address_space(3))) v2i* lptr;   // LDS side
#endif

__device__ __forceinline__ void wait_async0() {
#if defined(__HIP_DEVICE_COMPILE__)
#if __has_builtin(__builtin_amdgcn_s_wait_asynccnt)
    __builtin_amdgcn_s_wait_asynccnt(0);
#else
    asm volatile("s_wait_asynccnt 0x0" ::: "memory");
#endif
#endif
}

// ---------------------------------------------------------------------
// Kernel 1: time-varying complex FIR for f in [0, 96).
// One lane per frequency (96 = 3 wave32s), TT time steps per block with a
// register sliding window over spec (each spec value loaded exactly once
// per block; coefs streamed once, fully coalesced float2 loads).
// ---------------------------------------------------------------------
__global__ __launch_bounds__(96) void df_filter(const float* __restrict__ spec,
                                                const float* __restrict__ coefs,
                                                float* __restrict__ out) {
    const int f  = threadIdx.x;        // 0..95
    const int b  = blockIdx.y;         // 0..15
    const int t0 = blockIdx.x * TT;    // multiple of TT

    const f2* sp = (const f2*)spec  + (unsigned)(b * TLEN) * FTOT + f;
    f2*       op = (f2*)out         + (unsigned)(b * TLEN) * FTOT + f;
    const f2* cp = (const f2*)coefs + (unsigned)(b * FS) * TLEN * NFREQ + f;
    // tap i, time t lives at cp[(i*TLEN + t)*NFREQ]

    // prime sliding window: w[j] = spec(t0 - 4 + j), zero-padded before t=0
    f2 w[FS];
#pragma unroll
    for (int j = 0; j < FS; ++j) {
        const int tj = t0 - PADB + j;
        if (tj >= 0) w[j] = sp[(unsigned)tj * FTOT];
        else         { w[j].x = 0.f; w[j].y = 0.f; }
    }

    // warm the coef stream for this block (emits global_prefetch_b8)
#pragma unroll
    for (int i = 0; i < FS; ++i)
        __builtin_prefetch(&cp[((unsigned)(i * TLEN) + t0) * NFREQ], 0, 0);

#pragma unroll
    for (int s = 0; s < TT; ++s) {
        const int t = t0 + s;
        float ar = 0.f, ai = 0.f;
#pragma unroll
        for (int i = 0; i < FS; ++i) {
            const f2 c = cp[((unsigned)(i * TLEN) + t) * NFREQ];
            ar = fmaf(w[i].x,  c.x, ar);   // + sr*cr
            ar = fmaf(-w[i].y, c.y, ar);   // - si*ci
            ai = fmaf(w[i].x,  c.y, ai);   // + sr*ci
            ai = fmaf(w[i].y,  c.x, ai);   // + si*cr
        }
        f2 r; r.x = ar; r.y = ai;
        op[(unsigned)t * FTOT] = r;

        // slide the window
#pragma unroll
        for (int j = 0; j < FS - 1; ++j) w[j] = w[j + 1];
        if (s + 1 < TT) w[FS - 1] = sp[(unsigned)(t + 1) * FTOT];
    }
}

// ---------------------------------------------------------------------
// Kernel 2: passthrough copy for f in [96, 481), 8 bytes (one complex)
// per element. Uses the gfx1250 async global<->LDS DMA path (ASYNCcnt
// tracked, no VGPR round trip); plain coalesced float2 copy otherwise.
// ---------------------------------------------------------------------
__global__ __launch_bounds__(CPB_THREADS) void df_copy(const float* __restrict__ spec,
                                                       float* __restrict__ out) {
    const unsigned tid  = threadIdx.x;
    const unsigned base = blockIdx.x * (CPB_THREADS * CPB_K);

#if USE_ASYNC_COPY
    __shared__ v2i lbuf[CPB_THREADS * CPB_K];
    const v2i* in8 = (const v2i*)spec;
    v2i*       out8 = (v2i*)out;
    lptr lb = (lptr)lbuf;

    unsigned gidx[CPB_K];
#pragma unroll
    for (int k = 0; k < CPB_K; ++k) {
        const unsigned e   = base + (unsigned)k * CPB_THREADS + tid;
        const unsigned row = e / (unsigned)NPASS;
        const unsigned g   = row * (unsigned)FTOT + (e - row * (unsigned)NPASS) + (unsigned)NFREQ;
        gidx[k] = g;
        if (e < (unsigned)NCOPY) {
            __builtin_amdgcn_global_load_async_to_lds_b64(
                (gptr)(in8 + g), lb + ((unsigned)k * CPB_THREADS + tid), 0, 0);
        }
    }
    wait_async0();   // all async loads for this wave landed in LDS
#pragma unroll
    for (int k = 0; k < CPB_K; ++k) {
        const unsigned e = base + (unsigned)k * CPB_THREADS + tid;
        if (e < (unsigned)NCOPY) {
            __builtin_amdgcn_global_store_async_from_lds_b64(
                (gptr)(out8 + gidx[k]), lb + ((unsigned)k * CPB_THREADS + tid), 0, 0);
        }
    }
    wait_async0();   // drain async stores before wave end
#else
    const unsigned long long* in8  = (const unsigned long long*)spec;
    unsigned long long*       out8 = (unsigned long long*)out;
#pragma unroll
    for (int k = 0; k < CPB_K; ++k) {
        const unsigned e = base + (unsigned)k * CPB_THREADS + tid;
        if (e < (unsigned)NCOPY) {
            const unsigned row = e / (unsigned)NPASS;
            const unsigned g   = row * (unsigned)FTOT + (e - row * (unsigned)NPASS) + (unsigned)NFREQ;
            out8[g] = in8[g];
        }
    }
#endif
}

// ---------------------------------------------------------------------
extern "C" void kernel_launch(void* const* d_in, const int* in_sizes, int n_in,
                              void* d_out, int out_size, void* d_ws, size_t ws_size,
                              hipStream_t stream) {
    (void)in_sizes; (void)n_in; (void)out_size; (void)d_ws; (void)ws_size;
    const float* spec  = (const float*)d_in[0];
    const float* coefs = (const float*)d_in[1];
    float*       out   = (float*)d_out;

    dim3 gf(TLEN / TT, BATCH);           // 500 x 16 blocks, 96 threads each
    df_filter<<<gf, dim3(96), 0, stream>>>(spec, coefs, out);

    const unsigned span = CPB_THREADS * CPB_K;                 // 2048 elems/block
    const unsigned nblk = ((unsigned)NCOPY + span - 1) / span; // 12032 blocks
    df_copy<<<dim3(nblk), dim3(CPB_THREADS), 0, stream>>>(spec, out);
}